// DeChunkLayer_53584011985408
// MI455X (gfx1250) — compile-verified
//
#include <hip/hip_runtime.h>
#include <hip/hip_bf16.h>

// Problem constants from the reference
#define BB 4
#define LL 8192
#define DD 1024
#define BL (BB * LL)

// Two-level scan partitioning
#define SEGLEN 256
#define SEG_N (LL / SEGLEN)   // 32 segments per row
#define NCH (SEGLEN / 16)     // 16 chunks per segment

typedef __attribute__((ext_vector_type(2))) float v2f;
typedef __attribute__((ext_vector_type(8))) float v8f;
typedef __attribute__((ext_vector_type(4))) unsigned int u32x4;
typedef __attribute__((ext_vector_type(8))) int i32x8;
typedef __attribute__((ext_vector_type(4))) int i32x4;

// ---------------------------------------------------------------------------
// TDM: async-load one 16x16 fp32 tile (row stride DD) from global into LDS.
// D# group0: count=1 | lds_addr | global_addr[56:0] | type=2
// D# group1: data_size=4B, tensor_dim0/1=16, tile_dim0/1=16, dim0_stride=DD
// Groups 2/3 zero (2D tensor). Tracked by TENSORcnt.
// This toolchain exposes the 6-arg builtin: (g0, g1, g2, g3, g4, cpol).
// ---------------------------------------------------------------------------
__device__ __forceinline__ void tdm_load_tile16x16(unsigned lds_off,
                                                   unsigned long long gaddr) {
    u32x4 g0;
    g0.x = 1u;                                        // count = 1 descriptor
    g0.y = lds_off;                                   // LDS byte address
    g0.z = (unsigned)gaddr;                           // global_addr[31:0]
    g0.w = (unsigned)(gaddr >> 32) | (2u << 30);      // global_addr[56:32] | type=2
    i32x8 g1;
    g1[0] = 0x20000;                                  // data_size = 2 (4 bytes)
    g1[1] = (int)(16u << 16);                         // tensor_dim0 = 16
    g1[2] = (int)(16u << 16);                         // tensor_dim1 = 16
    g1[3] = (int)(16u << 16);                         // tile_dim0 = 16
    g1[4] = 16;                                       // tile_dim1 = 16, tile_dim2 = 0
    g1[5] = DD;                                       // tensor_dim0_stride = 1024
    g1[6] = 0;
    g1[7] = 0;
    i32x4 z4 = {0, 0, 0, 0};
    i32x8 z8 = {0, 0, 0, 0, 0, 0, 0, 0};
    __builtin_amdgcn_tensor_load_to_lds(g0, g1, z4, z4, z8, 0);
}

// ---------------------------------------------------------------------------
// Phase 0: per-row block scan over the boundary mask.
// ---------------------------------------------------------------------------
__global__ void dechunk_scan_rank(const float* __restrict__ bprob,
                                  const unsigned char* __restrict__ mask,
                                  float* __restrict__ p_sorted,
                                  int* __restrict__ rep_start,
                                  int* __restrict__ kb_arr) {
    const int b = blockIdx.x;
    const int tid = threadIdx.x;      // 1024 threads
    const int CH = LL / 1024;         // 8 elements per thread
    __shared__ int ssum[1024];

    const int base = tid * CH;
    unsigned char mloc[CH];
    int loc[CH];
    int run = 0;
    for (int i = 0; i < CH; ++i) {
        mloc[i] = mask[b * LL + base + i];
        run += mloc[i] ? 1 : 0;
        loc[i] = run;
    }
    ssum[tid] = run;
    __syncthreads();
    for (int off = 1; off < 1024; off <<= 1) {
        int add = (tid >= off) ? ssum[tid - off] : 0;
        __syncthreads();
        ssum[tid] += add;
        __syncthreads();
    }
    const int incl_thread = ssum[tid];
    const int excl = incl_thread - run;
    const int kb = ssum[1023];
    if (tid == 0) kb_arr[b] = kb;

    for (int i = 0; i < CH; ++i) {
        const int l = base + i;
        const int incl = excl + loc[i];
        const bool m = (mloc[i] != 0);
        const int rank = m ? (incl - 1) : (kb + (l - incl));
        float p = bprob[(size_t)(b * LL + l) * 2 + 1];
        p = fminf(fmaxf(p, 1e-4f), 1.0f - 1e-4f);
        p_sorted[b * LL + rank] = p;
        if (m) rep_start[b * LL + (incl - 1)] = l;
    }
}

// ---------------------------------------------------------------------------
// Phase 0b: per-position recurrence scalars (shared across all D channels).
// ---------------------------------------------------------------------------
__global__ void dechunk_scalars(const float* __restrict__ p_sorted,
                                float* __restrict__ a_arr,
                                float* __restrict__ coef_arr,
                                float* __restrict__ rdt_arr) {
    const int idx = blockIdx.x * blockDim.x + threadIdx.x;
    if (idx >= BL) return;
    const float p = p_sorted[idx];
    const float dt32 = __bfloat162float(__float2bfloat16(logf(1.0f / (1.0f - p))));
    const float b32 = __bfloat162float(__float2bfloat16(p));
    a_arr[idx] = expf(-dt32);
    coef_arr[idx] = dt32 * b32;
    rdt_arr[idx] = 1.0f / dt32;   // full IEEE divide, once per (b,t)
}

// ---------------------------------------------------------------------------
// Per-segment total decay product (scalar per (b,seg))
// ---------------------------------------------------------------------------
__global__ void __launch_bounds__(32)
dechunk_segprod(const float* __restrict__ a_arr, float* __restrict__ aseg) {
    const int s = blockIdx.x;
    const int b = blockIdx.y;
    const int lane = threadIdx.x;
    const int per = SEGLEN / 32;
    float p = 1.0f;
    const int base = b * LL + s * SEGLEN + lane * per;
    for (int i = 0; i < per; ++i) p *= a_arr[base + i];
    for (int off = 16; off >= 1; off >>= 1) p *= __shfl_xor(p, off, 32);
    if (lane == 0) aseg[b * SEG_N + s] = p;
}

// ---------------------------------------------------------------------------
// Pass B: scan the segment aggregates -> incoming state per segment.
// ---------------------------------------------------------------------------
__global__ void dechunk_carry(const float* __restrict__ aseg,
                              const float* __restrict__ hend,
                              float* __restrict__ hcarry) {
    const int d = blockIdx.x * blockDim.x + threadIdx.x;
    const int b = blockIdx.y;
    float H = 0.0f;
    for (int s = 0; s < SEG_N; ++s) {
        hcarry[((size_t)(b * SEG_N + s)) * DD + d] = H;
        if (s < SEG_N - 1)
            H = fmaf(aseg[b * SEG_N + s], H, hend[((size_t)(b * SEG_N + s)) * DD + d]);
    }
}

// ---------------------------------------------------------------------------
// Chunked linear scan via V_WMMA_F32_16X16X4_F32 over one 256-step segment,
// with the hidden tiles streamed by the Tensor Data Mover (quad-buffered LDS,
// issue depth 3 => ~3KB in flight per wave; x8192 waves covers the HBM
// bandwidth-delay product). FINAL=false: emit segment end state only.
// FINAL=true: start from hcarry and write replicated fp32(bf16(h)) output.
// ---------------------------------------------------------------------------
#define TSTR 20

template <bool FINAL>
__global__ void __launch_bounds__(32)
dechunk_seg_scan(const float* __restrict__ hidden,
                 const float* __restrict__ a_arr,
                 const float* __restrict__ coef_arr,
                 const float* __restrict__ rdt_arr,
                 const int* __restrict__ rep_start,
                 const int* __restrict__ kb_arr,
                 const float* __restrict__ hcarry,
                 float* __restrict__ hend,
                 float* __restrict__ out) {
    const int b = blockIdx.z;
    const int seg = blockIdx.y;
    const int d0 = blockIdx.x * 16;
    const int lane = threadIdx.x;
    const int m = lane & 15;
    const int h = lane >> 4;

    __shared__ float sA[16];
    __shared__ float sCoef[16];
    __shared__ float sRdt[16];
    __shared__ float sT[16 * TSTR];
    __shared__ int sRep[17];
    __shared__ float sX[4 * 256];   // 4 x (16t x 16d) fp32 tiles

    const int kb = FINAL ? kb_arr[b] : 0;
    float bcarry = FINAL
        ? hcarry[((size_t)(b * SEG_N + seg)) * DD + d0 + m]
        : 0.0f;

    float* out_col = out + ((size_t)b * LL) * DD + d0 + m;

    // TDM base addresses: LDS offset = low 32 bits of the flat LDS pointer
    const unsigned ldsX = (unsigned)(uintptr_t)(void*)sX;
    const unsigned long long gbase = (unsigned long long)(uintptr_t)(
        hidden + ((size_t)(b * LL) + (size_t)seg * SEGLEN) * DD + d0);

    // Prologue: prime 3 tiles
    for (int i = 0; i < 3; ++i)
        tdm_load_tile16x16(ldsX + (unsigned)i * 1024u,
                           gbase + (unsigned long long)i * (16ull * DD * 4ull));

    for (int cc = 0; cc < NCH; ++cc) {
        const int l0 = seg * SEGLEN + cc * 16;

        // Keep the pipeline full: issue tile cc+3 while computing cc
        if (cc + 3 < NCH) {
            tdm_load_tile16x16(ldsX + (unsigned)((cc + 3) & 3) * 1024u,
                               gbase + (unsigned long long)(cc + 3) * (16ull * DD * 4ull));
        }

        if (lane < 16) {
            const int g = b * LL + l0 + lane;
            sA[lane] = a_arr[g];
            sCoef[lane] = coef_arr[g];
            sRdt[lane] = rdt_arr[g];
            if (FINAL) sRep[lane] = (l0 + lane < kb) ? rep_start[g] : LL;
        } else if (FINAL && lane == 16) {
            sRep[16] = (l0 + 16 < kb) ? rep_start[b * LL + l0 + 16] : LL;
        }
        __syncthreads();

        // T[i][k] = coef_k * prod_{j=k+1..i} a_j ; col16 = inclusive cumprod a
        if (lane < 16) {
            const int i = lane;
            float* row = &sT[i * TSTR];
            float r = 1.0f;
            row[i] = sCoef[i];
            for (int k = i - 1; k >= 0; --k) {
                r *= sA[k + 1];
                row[k] = sCoef[k] * r;
            }
            row[16] = r * sA[0];
            for (int k = i + 1; k < 16; ++k) row[k] = 0.0f;
            row[17] = 0.0f; row[18] = 0.0f; row[19] = 0.0f;
        }
        __syncthreads();

        // Wait for tile cc (tensor ops complete in order per wave)
        if (cc + 3 < NCH)       __builtin_amdgcn_s_wait_tensorcnt(3);
        else if (cc + 3 == NCH) __builtin_amdgcn_s_wait_tensorcnt(2);
        else if (cc + 2 == NCH) __builtin_amdgcn_s_wait_tensorcnt(1);
        else                    __builtin_amdgcn_s_wait_tensorcnt(0);
        __syncthreads();

        const float* xt = &sX[(cc & 3) * 256];
        v8f C = {0.0f, 0.0f, 0.0f, 0.0f, 0.0f, 0.0f, 0.0f, 0.0f};

        for (int s = 0; s < 4; ++s) {
            const int kk = 4 * s + 2 * h;
            v2f Aop;
            Aop.x = sT[m * TSTR + kk];
            Aop.y = sT[m * TSTR + kk + 1];
            v2f Bop;
            Bop.x = __bfloat162float(__float2bfloat16(xt[kk * 16 + m] * sRdt[kk]));
            Bop.y = __bfloat162float(__float2bfloat16(xt[(kk + 1) * 16 + m] * sRdt[kk + 1]));
            C = __builtin_amdgcn_wmma_f32_16x16x4_f32(
                    false, Aop, false, Bop, (short)0, C, false, false);
        }
        {   // carry slab (K=16..19; only K=16 non-zero)
            v2f Aop;
            Aop.x = sT[m * TSTR + 16 + 2 * h];
            Aop.y = sT[m * TSTR + 17 + 2 * h];
            v2f Bop;
            Bop.x = (h == 0) ? bcarry : 0.0f;
            Bop.y = 0.0f;
            C = __builtin_amdgcn_wmma_f32_16x16x4_f32(
                    false, Aop, false, Bop, (short)0, C, false, false);
        }

        // New carry: C row 15 lives in VGPR 7 of lanes 16..31
        bcarry = __shfl(C[7], 16 + m, 32);

        if (FINAL) {
            // lane holds column d0+m, rows h*8+r; replicate across segment
            for (int r = 0; r < 8; ++r) {
                const int trow = h * 8 + r;
                const int tglob = l0 + trow;
                if (tglob < kb) {
                    const int seg_s = sRep[trow];
                    const int seg_e = sRep[trow + 1];
                    const float val = __bfloat162float(__float2bfloat16(C[r]));
                    for (int lq = seg_s; lq < seg_e; ++lq) {
                        out_col[(size_t)lq * DD] = val;
                    }
                }
            }
        }
        __syncthreads();
    }

    if (!FINAL) {
        if (lane < 16)
            hend[((size_t)(b * SEG_N + seg)) * DD + d0 + lane] = bcarry;
    }
}

// ---------------------------------------------------------------------------
extern "C" void kernel_launch(void* const* d_in, const int* in_sizes, int n_in,
                              void* d_out, int out_size, void* d_ws, size_t ws_size,
                              hipStream_t stream) {
    const float* hidden = (const float*)d_in[0];                // [B,L,D] fp32
    const float* bprob = (const float*)d_in[1];                 // [B,L,2] fp32
    const unsigned char* mask = (const unsigned char*)d_in[2];  // [B,L] bool
    float* out = (float*)d_out;                                 // [B,L,D] fp32

    float* ws = (float*)d_ws;
    float* a_arr = ws + 0 * BL;
    float* coef_arr = ws + 1 * BL;
    float* rdt_arr = ws + 2 * BL;
    float* p_sorted = ws + 3 * BL;
    int* rep_start = (int*)(ws + 4 * BL);
    int* kb_arr = (int*)(ws + 5 * BL);                 // 16 ints
    float* aseg = ws + 5 * BL + 16;                    // B*SEG_N floats
    float* hend = ws + 5 * BL + 16 + BB * SEG_N;       // B*SEG_N*D floats
    float* hcarry = hend + (size_t)BB * SEG_N * DD;    // B*SEG_N*D floats

    dechunk_scan_rank<<<BB, 1024, 0, stream>>>(bprob, mask, p_sorted, rep_start, kb_arr);
    dechunk_scalars<<<(BL + 255) / 256, 256, 0, stream>>>(p_sorted, a_arr, coef_arr, rdt_arr);
    dechunk_segprod<<<dim3(SEG_N, BB), 32, 0, stream>>>(a_arr, aseg);

    // Pass A: per-segment end states (segments 0..S-2 feed the carry scan)
    dechunk_seg_scan<false><<<dim3(DD / 16, SEG_N - 1, BB), 32, 0, stream>>>(
        hidden, a_arr, coef_arr, rdt_arr, rep_start, kb_arr, hcarry, hend, out);
    // Pass B: scan segment aggregates -> incoming carry per segment
    dechunk_carry<<<dim3(DD / 256, BB), 256, 0, stream>>>(aseg, hend, hcarry);
    // Pass C: full chunked WMMA scan with carries + replicated stores
    dechunk_seg_scan<true><<<dim3(DD / 16, SEG_N, BB), 32, 0, stream>>>(
        hidden, a_arr, coef_arr, rdt_arr, rep_start, kb_arr, hcarry, hend, out);

    (void)in_sizes; (void)n_in; (void)out_size; (void)ws_size;
}